// GNN_38053410243243
// MI455X (gfx1250) — compile-verified
//
#include <hip/hip_runtime.h>

typedef __attribute__((ext_vector_type(2))) float v2f;
typedef __attribute__((ext_vector_type(8))) float v8f;

#define NVV   100000
#define NCC   50000
#define DVV   8
#define DCC   16
#define DIN   32
#define DH    128
#define DCAT  (2*DH + DIN)

// Pre-swizzle W (K x 128 row-major) into per-lane WMMA B-fragment order:
// Wsw[step][lane][nt*2 + j] = W[step*4 + (lane>>4)*2 + j][nt*16 + (lane&15)]
// so each lane reads its 8 B-fragments for one K-step as 4 consecutive b128s.
__global__ __launch_bounds__(256) void swizzleW(
    const float* __restrict__ W, float* __restrict__ Wsw, int K)
{
  int i = blockIdx.x * blockDim.x + threadIdx.x;
  int total = (K >> 2) * 512;
  if (i >= total) return;
  int step = i >> 9;
  int r    = i & 511;
  int lane = r >> 4;
  int e    = r & 15;          // nt*2 + j
  int nt   = e >> 1;
  int j    = e & 1;
  int krow = step * 4 + ((lane >> 4) << 1) + j;
  int col  = nt * 16 + (lane & 15);
  Wsw[i] = W[(size_t)krow * DH + col];
}

// out[M x 128] = A1[M x K1] @ W[0:K1,:] + A2 @ W[K1:K1+K2,:] + A3 @ W[...,:] + bias
// Wsw is the swizzled (K1+K2+K3) x 128 weight (see swizzleW). Each Ai row-major,
// leading dim == Ki. Block = 128 threads = 4 waves; each wave: 32 rows x 128 cols.
__global__ __launch_bounds__(128) void gemm3_wmma(
    const float* __restrict__ A1, int K1,
    const float* __restrict__ A2, int K2,
    const float* __restrict__ A3, int K3,
    const float* __restrict__ Wsw, const float* __restrict__ bias,
    float* __restrict__ out, int M)
{
  const int wave = threadIdx.x >> 5;
  const int lane = threadIdx.x & 31;
  const int l16  = lane & 15;
  const int kb   = (lane >> 4) << 1;          // 0 (lanes 0-15) or 2 (lanes 16-31)
  const int row0 = blockIdx.x * 128 + wave * 32;

  int rA = row0 + l16;                         // rows for fragment set 0
  int rB = row0 + 16 + l16;                    // rows for fragment set 1
  int rAc = rA < M ? rA : (M - 1);             // clamp for safe tail loads
  int rBc = rB < M ? rB : (M - 1);

  v8f acc0[8], acc1[8];
#pragma unroll
  for (int nt = 0; nt < 8; ++nt) {
    float bv = bias[nt * 16 + l16];            // N depends only on lane
    v8f t;
#pragma unroll
    for (int r = 0; r < 8; ++r) t[r] = bv;
    acc0[nt] = t;
    acc1[nt] = t;
  }

  const float* Aseg[3] = {A1, A2, A3};
  const int    Kseg[3] = {K1, K2, K3};
  int kglob = 0;
  for (int s = 0; s < 3; ++s) {
    const float* A = Aseg[s];
    const int K = Kseg[s];
    for (int k = 0; k < K; k += 4) {
      // A fragments: lane half selects K pair {k,k+1} vs {k+2,k+3}
      v2f a0 = *(const v2f*)(A + (size_t)rAc * K + k + kb);
      v2f a1 = *(const v2f*)(A + (size_t)rBc * K + k + kb);

      // B fragments: 4 x b128 per lane from swizzled W
      const float4* Ws =
          (const float4*)(Wsw + (size_t)((kglob + k) >> 2) * 512 + lane * 16);
      float4 w0 = Ws[0];
      float4 w1 = Ws[1];
      float4 w2 = Ws[2];
      float4 w3 = Ws[3];
      v2f b[8];
      b[0].x = w0.x; b[0].y = w0.y;  b[1].x = w0.z; b[1].y = w0.w;
      b[2].x = w1.x; b[2].y = w1.y;  b[3].x = w1.z; b[3].y = w1.w;
      b[4].x = w2.x; b[4].y = w2.y;  b[5].x = w2.z; b[5].y = w2.w;
      b[6].x = w3.x; b[6].y = w3.y;  b[7].x = w3.z; b[7].y = w3.w;

#pragma unroll
      for (int nt = 0; nt < 8; ++nt) {
        acc0[nt] = __builtin_amdgcn_wmma_f32_16x16x4_f32(
            false, a0, false, b[nt], (short)0, acc0[nt], false, false);
        acc1[nt] = __builtin_amdgcn_wmma_f32_16x16x4_f32(
            false, a1, false, b[nt], (short)0, acc1[nt], false, false);
      }
    }
    kglob += K;
  }

  // Store: C/D layout -> VGPR r holds M = r (lanes 0-15) or r+8 (lanes 16-31)
  const int mhalf = (lane >> 4) << 3;
#pragma unroll
  for (int r = 0; r < 8; ++r) {
    int m0 = row0 + r + mhalf;
    if (m0 < M) {
      float* o = out + (size_t)m0 * DH + l16;
#pragma unroll
      for (int nt = 0; nt < 8; ++nt) o[nt * 16] = acc0[nt][r];
    }
    int m1 = row0 + 16 + r + mhalf;
    if (m1 < M) {
      float* o = out + (size_t)m1 * DH + l16;
#pragma unroll
      for (int nt = 0; nt < 8; ++nt) o[nt * 16] = acc1[nt][r];
    }
  }
}

// agg[i,:] = sum_{j<deg} h[idx[i,j],:]   (one wave per row, float4 per lane)
__global__ __launch_bounds__(256) void gather_sum(
    const float* __restrict__ h, const int* __restrict__ idx,
    int deg, int nrows, float* __restrict__ agg)
{
  int wid  = blockIdx.x * (blockDim.x >> 5) + (threadIdx.x >> 5);
  int lane = threadIdx.x & 31;
  if (wid >= nrows) return;
  const int* ri = idx + (size_t)wid * deg;
  float4 s = make_float4(0.f, 0.f, 0.f, 0.f);
  for (int j = 0; j < deg; ++j) {
    int nb = ri[j];
    float4 t = ((const float4*)(h + (size_t)nb * DH))[lane];
    s.x += t.x; s.y += t.y; s.z += t.z; s.w += t.w;
  }
  ((float4*)(agg + (size_t)wid * DH))[lane] = s;
}

__global__ void colsum_zero(float* __restrict__ g)
{
  if (threadIdx.x < DH) g[threadIdx.x] = 0.f;
}

__global__ __launch_bounds__(128) void colsum(
    const float* __restrict__ v, float* __restrict__ g, int M)
{
  float s = 0.f;
  const int n = threadIdx.x;
  for (int i = blockIdx.x; i < M; i += gridDim.x)
    s += v[(size_t)i * DH + n];
  atomicAdd(&g[n], s);
}

// s = dot(g, W_q[0:128,0]) + b_q
__global__ __launch_bounds__(128) void qscalar(
    const float* __restrict__ g, const float* __restrict__ Wq,
    const float* __restrict__ bq, float* __restrict__ sout)
{
  __shared__ float red[DH];
  red[threadIdx.x] = g[threadIdx.x] * Wq[threadIdx.x];
  __syncthreads();
  for (int off = 64; off > 0; off >>= 1) {
    if (threadIdx.x < off) red[threadIdx.x] += red[threadIdx.x + off];
    __syncthreads();
  }
  if (threadIdx.x == 0) sout[0] = red[0] + bq[0];
}

// out[i] = dot(v[i,:], W_q[128:256,0]) + s   (one wave per row)
__global__ __launch_bounds__(256) void qfinal(
    const float* __restrict__ v, const float* __restrict__ Wq,
    const float* __restrict__ s, float* __restrict__ out, int M)
{
  int wid  = blockIdx.x * (blockDim.x >> 5) + (threadIdx.x >> 5);
  int lane = threadIdx.x & 31;
  if (wid >= M) return;
  float4 a = ((const float4*)(v + (size_t)wid * DH))[lane];
  float4 b = ((const float4*)(Wq + DH))[lane];
  float p = a.x * b.x + a.y * b.y + a.z * b.z + a.w * b.w;
  for (int off = 16; off > 0; off >>= 1)
    p += __shfl_xor(p, off, 32);
  if (lane == 0) out[wid] = p + s[0];
}

extern "C" void kernel_launch(void* const* d_in, const int* in_sizes, int n_in,
                              void* d_out, int out_size, void* d_ws, size_t ws_size,
                              hipStream_t stream) {
  const float* x    = (const float*)d_in[0];
  const int*   vci  = (const int*)d_in[1];   // (NV, DV) var -> constr
  const int*   cvi  = (const int*)d_in[2];   // (NC, DC) constr -> var
  const float* Wiv  = (const float*)d_in[3];
  const float* biv  = (const float*)d_in[4];
  const float* Wic  = (const float*)d_in[5];
  const float* bic  = (const float*)d_in[6];
  const float* Wvar = (const float*)d_in[7];
  const float* bvar = (const float*)d_in[8];
  const float* Wcon = (const float*)d_in[9];
  const float* bcon = (const float*)d_in[10];
  const float* Wq   = (const float*)d_in[11];
  const float* bq   = (const float*)d_in[12];

  const float* xv = x;
  const float* xc = x + (size_t)NVV * DIN;

  float* ws    = (float*)d_ws;
  float* v     = ws;                               // NV*DH
  float* c     = v    + (size_t)NVV * DH;          // NC*DH
  float* aggv  = c    + (size_t)NCC * DH;          // NV*DH
  float* aggc  = aggv + (size_t)NVV * DH;          // NC*DH
  float* g     = aggc + (size_t)NCC * DH;          // DH
  float* sp    = g + DH;                           // 1 (+pad to 128-float align)
  float* wswV  = g + 2 * DH;                       // DCAT*DH swizzled W_var
  float* wswC  = wswV + (size_t)DCAT * DH;         // DCAT*DH swizzled W_con
  float* wswIV = wswC + (size_t)DCAT * DH;         // DIN*DH swizzled W_init_v
  float* wswIC = wswIV + (size_t)DIN * DH;         // DIN*DH swizzled W_init_c

  float* out = (float*)d_out;

  const int gemmV = (NVV + 127) / 128;
  const int gemmC = (NCC + 127) / 128;

  // one-shot weight swizzles (tiny; amortized against ~24 GFLOP of GEMM)
  {
    int tc = (DCAT / 4) * 512;
    swizzleW<<<(tc + 255) / 256, 256, 0, stream>>>(Wvar, wswV, DCAT);
    swizzleW<<<(tc + 255) / 256, 256, 0, stream>>>(Wcon, wswC, DCAT);
    int ti = (DIN / 4) * 512;
    swizzleW<<<(ti + 255) / 256, 256, 0, stream>>>(Wiv, wswIV, DIN);
    swizzleW<<<(ti + 255) / 256, 256, 0, stream>>>(Wic, wswIC, DIN);
  }

  // init: v = xv @ W_init_v + b ; c = xc @ W_init_c + b
  gemm3_wmma<<<gemmV, 128, 0, stream>>>(xv, DIN, nullptr, 0, nullptr, 0,
                                        wswIV, biv, v, NVV);
  gemm3_wmma<<<gemmC, 128, 0, stream>>>(xc, DIN, nullptr, 0, nullptr, 0,
                                        wswIC, bic, c, NCC);

  for (int layer = 0; layer < 2; ++layer) {
    // gathers from OLD v,c
    gather_sum<<<(NCC + 7) / 8, 256, 0, stream>>>(v, cvi, DCC, NCC, aggc);
    gather_sum<<<(NVV + 7) / 8, 256, 0, stream>>>(c, vci, DVV, NVV, aggv);
    // c = [agg_c, c, xc] @ W_con + b_con   (in-place: row i reads only row i of c)
    gemm3_wmma<<<gemmC, 128, 0, stream>>>(aggc, DH, c, DH, xc, DIN,
                                          wswC, bcon, c, NCC);
    // v = [agg_v, v, xv] @ W_var + b_var
    gemm3_wmma<<<gemmV, 128, 0, stream>>>(aggv, DH, v, DH, xv, DIN,
                                          wswV, bvar, v, NVV);
  }

  // g = column sum of v; s = dot(g, Wq[0:128]) + bq; out = v @ Wq[128:256] + s
  colsum_zero<<<1, 128, 0, stream>>>(g);
  colsum<<<512, 128, 0, stream>>>(v, g, NVV);
  qscalar<<<1, 128, 0, stream>>>(g, Wq, bq, sp);
  qfinal<<<(NVV + 7) / 8, 256, 0, stream>>>(v, Wq, sp, out, NVV);
}